// ImprovedStrictPhysicsModel_13469017440460
// MI455X (gfx1250) — compile-verified
//
#include <hip/hip_runtime.h>
#include <stdint.h>

typedef __attribute__((ext_vector_type(2))) float v2f;
typedef __attribute__((ext_vector_type(8))) float v8f;

#define T_STEPS 32768
#define NZ 64
#define DIM 97

// ---------------------------------------------------------------------------
// 16-row GEMM helper using V_WMMA_F32_16X16X4_F32.
//   out[16][NOUT] = in[16][K] @ W^T + bias     (W is [NOUT][K] row-major)
// A operand (16x4 f32): lanes 0-15 hold M=lane, VGPR0 = K=k (+2 for half 1),
// VGPR1 = K=k+1 (+2 for half 1).  B operand (4x16): lanes hold N, K split the
// same way.  C/D: VGPR i -> row i (lanes 0-15) / row 8+i (lanes 16-31).
// Called by all 256 threads; wave-uniform loop bounds keep EXEC all-ones.
// ---------------------------------------------------------------------------
__device__ __forceinline__ void gemm16(const float* __restrict__ W,
                                       const float* __restrict__ bias,
                                       const float* in, int ldin,
                                       float* out, int ldout,
                                       int K, int NOUT, int tid) {
  const int wave = tid >> 5;
  const int lane = tid & 31;
  const int half = lane >> 4;
  const int l15  = lane & 15;
  for (int n0 = wave * 16; n0 < NOUT; n0 += 16 * 8) {
    v8f acc = {};
    const float* wrow = W + (size_t)(n0 + l15) * K;
    const float* arow = in + l15 * ldin;
    for (int k = 0; k < K; k += 4) {
      v2f a = *(const v2f*)(arow + k + 2 * half);   // ldin even -> 8B aligned
      v2f b = *(const v2f*)(wrow + k + 2 * half);   // K multiple of 4 -> aligned
      acc = __builtin_amdgcn_wmma_f32_16x16x4_f32(false, a, false, b,
                                                  (short)0, acc, false, false);
    }
    const float bn = bias[n0 + l15];
#pragma unroll
    for (int i = 0; i < 8; ++i) {
      const int r = half * 8 + i;
      out[r * ldout + n0 + l15] = acc[i] + bn;
    }
  }
}

// In-place LayerNorm (+optional ReLU) over 16 rows of width `width`.
// 16 threads per row; deterministic reduction order.
__device__ __forceinline__ void ln_rows(float* buf, int ld, int width,
                                        const float* __restrict__ g,
                                        const float* __restrict__ bb,
                                        float* red, int tid, bool relu) {
  const int r = tid >> 4, c = tid & 15;
  const int per = width >> 4;
  float s = 0.f, ss = 0.f;
  for (int i = 0; i < per; ++i) {
    float v = buf[r * ld + c * per + i];
    s += v; ss += v * v;
  }
  red[r * 32 + c] = s;
  red[r * 32 + 16 + c] = ss;
  __syncthreads();
  float S = 0.f, SS = 0.f;
  for (int i = 0; i < 16; ++i) { S += red[r * 32 + i]; SS += red[r * 32 + 16 + i]; }
  const float m = S / (float)width;
  const float var = SS / (float)width - m * m;
  const float inv = rsqrtf(fmaxf(var, 0.f) + 1e-5f);
  for (int i = 0; i < per; ++i) {
    const int j = c * per + i;
    float v = (buf[r * ld + j] - m) * inv * g[j] + bb[j];
    if (relu) v = fmaxf(v, 0.f);
    buf[r * ld + j] = v;
  }
  __syncthreads();
}

// ---------------------------------------------------------------------------
// K0: adjacency mask + person_attrs moments
// ---------------------------------------------------------------------------
__global__ void setup_kernel(const float* __restrict__ pa,
                             const int* __restrict__ ei,
                             unsigned char* __restrict__ Am,
                             float* __restrict__ stats) {
  const int tid = threadIdx.x;
  for (int i = tid; i < NZ * NZ; i += 256) Am[i] = 0;
  __syncthreads();
  if (tid == 0) {
    for (int e = 0; e < 256; ++e) {
      int u = ei[e], v = ei[256 + e];
      Am[u * NZ + v] = 1;
      Am[v * NZ + u] = 1;
    }
    for (int z = 0; z < NZ; ++z) Am[z * NZ + z] = 1;
    float s = 0.f, ss = 0.f;
    for (int i = 0; i < 32; ++i) { float x = pa[i]; s += x; ss += x * x; }
    stats[0] = s;
    stats[1] = ss;
  }
}

// ---------------------------------------------------------------------------
// K1: for every (t, candidate-zone z), compute next[t][z] = masked argmax.
// Layer 1 shared across all 64 zones of a timestep via rank-1 correction.
// ---------------------------------------------------------------------------
__global__ __launch_bounds__(256) void phase1_kernel(
    const float* __restrict__ pa, const float* __restrict__ times,
    const float* __restrict__ g0, const float* __restrict__ b0,
    const float* __restrict__ W1, const float* __restrict__ b1,
    const float* __restrict__ g1, const float* __restrict__ be1,
    const float* __restrict__ W2, const float* __restrict__ b2,
    const float* __restrict__ g2, const float* __restrict__ be2,
    const float* __restrict__ W3, const float* __restrict__ b3,
    const float* __restrict__ g3, const float* __restrict__ be3,
    const float* __restrict__ W4, const float* __restrict__ b4,
    const unsigned char* __restrict__ Am, const float* __restrict__ stats,
    unsigned char* __restrict__ next) {
  __shared__ float xn0[104];
  __shared__ float base[256];
  __shared__ float h1[16][258];
  __shared__ float h2[16][132];
  __shared__ float h3[16][68];
  __shared__ float raw[16][68];
  __shared__ float red[512];
  __shared__ float sh_m, sh_inv;

  const int tid = threadIdx.x;
  const float S = stats[0], Ssq = stats[1];

  for (int ti = 0; ti < 8; ++ti) {
    const int t = blockIdx.x * 8 + ti;
    const float tv = times[t];
    if (tid == 0) {
      float m = (S + tv + 1.0f) / 97.0f;
      float var = (Ssq + tv * tv + 1.0f) / 97.0f - m * m;
      sh_m = m;
      sh_inv = rsqrtf(var + 1e-5f);
    }
    __syncthreads();
    const float m0 = sh_m, inv0 = sh_inv;
    if (tid < DIM) {
      float x = (tid < 32) ? pa[tid] : ((tid == 32) ? tv : 0.0f);
      xn0[tid] = g0[tid] * (x - m0) * inv0 + b0[tid];
    }
    __syncthreads();
    {  // base(t) = W1 @ xn0 + b1 (one output per thread)
      float acc = b1[tid];
      const float* wr = W1 + tid * DIM;
      for (int i = 0; i < DIM; ++i) acc += wr[i] * xn0[i];
      base[tid] = acc;
    }
    __syncthreads();

    for (int p = 0; p < 4; ++p) {
      const int z0 = p * 16;
      const int r = tid >> 4, c = tid & 15;
      const int z = z0 + r;
      const float cz = g0[33 + z] * inv0;  // rank-1 one-hot correction scale
      for (int i = 0; i < 16; ++i) {
        const int j = c * 16 + i;
        h1[r][j] = base[j] + cz * W1[j * DIM + 33 + z];
      }
      __syncthreads();
      ln_rows(&h1[0][0], 258, 256, g1, be1, red, tid, true);
      gemm16(W2, b2, &h1[0][0], 258, &h2[0][0], 132, 256, 128, tid);
      __syncthreads();
      ln_rows(&h2[0][0], 132, 128, g2, be2, red, tid, true);
      gemm16(W3, b3, &h2[0][0], 132, &h3[0][0], 68, 128, 64, tid);
      __syncthreads();
      ln_rows(&h3[0][0], 68, 64, g3, be3, red, tid, true);
      gemm16(W4, b4, &h3[0][0], 68, &raw[0][0], 68, 64, 64, tid);
      __syncthreads();
      if (tid < 16) {
        const int zz = z0 + tid;
        const unsigned char* mrow = Am + zz * NZ;
        float best = -3.4e38f;
        int bi = 0;
        for (int j = 0; j < NZ; ++j) {  // first-index tie-break like jnp.argmax
          float v = raw[tid][j];
          if (mrow[j] && v > best) { best = v; bi = j; }
        }
        next[(size_t)t * NZ + zz] = (unsigned char)bi;
      }
      __syncthreads();
    }
  }
}

// ---------------------------------------------------------------------------
// K2a: per-chunk transition maps (function composition over 512 steps)
// ---------------------------------------------------------------------------
__global__ void chunkmap_kernel(const unsigned char* __restrict__ next,
                                unsigned char* __restrict__ cmap) {
  const int c = blockIdx.x;
  const int e = threadIdx.x;  // 64 threads = 64 entry zones
  int cur = e;
  const int tbase = c * 512;
  for (int i = 0; i < 512; ++i) cur = next[(size_t)(tbase + i) * NZ + cur];
  cmap[c * NZ + e] = (unsigned char)cur;
}

// K2b+c: prefix-compose chunk maps, then replay chunks to emit zone[t]
__global__ void scan_kernel(const unsigned char* __restrict__ next,
                            const unsigned char* __restrict__ cmap,
                            int* __restrict__ zone) {
  __shared__ int start[64];
  const int tid = threadIdx.x;  // 64 threads = 64 chunks
  if (tid == 0) {
    int cur = 0;  // initial zone is 0
    for (int c = 0; c < 64; ++c) {
      start[c] = cur;
      cur = cmap[c * NZ + cur];
    }
  }
  __syncthreads();
  int cur = start[tid];
  const int tbase = tid * 512;
  for (int i = 0; i < 512; ++i) {
    zone[tbase + i] = cur;
    cur = next[(size_t)(tbase + i) * NZ + cur];
  }
}

// ---------------------------------------------------------------------------
// K3: with zone[t] known, recompute raw(t, zone_t) for 16 t's per block and
// write masked logits.
// ---------------------------------------------------------------------------
__global__ __launch_bounds__(256) void phase3_kernel(
    const float* __restrict__ pa, const float* __restrict__ times,
    const float* __restrict__ g0, const float* __restrict__ b0,
    const float* __restrict__ W1, const float* __restrict__ b1,
    const float* __restrict__ g1, const float* __restrict__ be1,
    const float* __restrict__ W2, const float* __restrict__ b2,
    const float* __restrict__ g2, const float* __restrict__ be2,
    const float* __restrict__ W3, const float* __restrict__ b3,
    const float* __restrict__ g3, const float* __restrict__ be3,
    const float* __restrict__ W4, const float* __restrict__ b4,
    const unsigned char* __restrict__ Am, const float* __restrict__ stats,
    const int* __restrict__ zone, float* __restrict__ out) {
  __shared__ float xn0[16][100];
  __shared__ float h1[16][258];
  __shared__ float h2[16][132];
  __shared__ float h3[16][68];
  __shared__ float raw[16][68];
  __shared__ float red[512];
  __shared__ float tvs[16], ms[16], invs[16];
  __shared__ int zs[16];

  const int tid = threadIdx.x;
  const int t0 = blockIdx.x * 16;
  const float S = stats[0], Ssq = stats[1];

  if (tid < 16) {
    const float tv = times[t0 + tid];
    tvs[tid] = tv;
    float m = (S + tv + 1.0f) / 97.0f;
    float var = (Ssq + tv * tv + 1.0f) / 97.0f - m * m;
    ms[tid] = m;
    invs[tid] = rsqrtf(var + 1e-5f);
    zs[tid] = zone[t0 + tid];
  }
  __syncthreads();

  const int r = tid >> 4, c = tid & 15;
  for (int i = c; i < DIM; i += 16) {
    float x = (i < 32) ? pa[i] : ((i == 32) ? tvs[r] : 0.0f);
    xn0[r][i] = g0[i] * (x - ms[r]) * invs[r] + b0[i];
  }
  __syncthreads();

  const int z = zs[r];
  const float cz = g0[33 + z] * invs[r];
  for (int jj = 0; jj < 16; ++jj) {
    const int j = c * 16 + jj;
    const float* wr = W1 + j * DIM;
    float acc = b1[j] + cz * wr[33 + z];
    for (int i = 0; i < DIM; ++i) acc += wr[i] * xn0[r][i];
    h1[r][j] = acc;
  }
  __syncthreads();
  ln_rows(&h1[0][0], 258, 256, g1, be1, red, tid, true);
  gemm16(W2, b2, &h1[0][0], 258, &h2[0][0], 132, 256, 128, tid);
  __syncthreads();
  ln_rows(&h2[0][0], 132, 128, g2, be2, red, tid, true);
  gemm16(W3, b3, &h2[0][0], 132, &h3[0][0], 68, 128, 64, tid);
  __syncthreads();
  ln_rows(&h3[0][0], 68, 64, g3, be3, red, tid, true);
  gemm16(W4, b4, &h3[0][0], 68, &raw[0][0], 68, 64, 64, tid);
  __syncthreads();

  const unsigned char* mrow = Am + z * NZ;
  for (int jj = 0; jj < 4; ++jj) {
    const int j = c * 4 + jj;
    out[(size_t)(t0 + r) * NZ + j] = mrow[j] ? raw[r][j] : -50.0f;
  }
}

// ---------------------------------------------------------------------------
extern "C" void kernel_launch(void* const* d_in, const int* in_sizes, int n_in,
                              void* d_out, int out_size, void* d_ws, size_t ws_size,
                              hipStream_t stream) {
  const float* pa    = (const float*)d_in[0];
  const float* times = (const float*)d_in[1];
  // d_in[2] = zone_features (unused by the reference)
  const int*   ei    = (const int*)d_in[3];
  const float* g0  = (const float*)d_in[4];
  const float* b0  = (const float*)d_in[5];
  const float* W1  = (const float*)d_in[6];
  const float* b1  = (const float*)d_in[7];
  const float* g1  = (const float*)d_in[8];
  const float* be1 = (const float*)d_in[9];
  const float* W2  = (const float*)d_in[10];
  const float* b2  = (const float*)d_in[11];
  const float* g2  = (const float*)d_in[12];
  const float* be2 = (const float*)d_in[13];
  const float* W3  = (const float*)d_in[14];
  const float* b3  = (const float*)d_in[15];
  const float* g3  = (const float*)d_in[16];
  const float* be3 = (const float*)d_in[17];
  const float* W4  = (const float*)d_in[18];
  const float* b4  = (const float*)d_in[19];
  float* out = (float*)d_out;

  // Workspace layout
  char* ws = (char*)d_ws;
  unsigned char* Am    = (unsigned char*)(ws + 0);         // 4096 B
  float*         stats = (float*)(ws + 4096);              // 8 B
  unsigned char* cmap  = (unsigned char*)(ws + 4352);      // 4096 B
  int*           zone  = (int*)(ws + 8448);                // 131072 B
  unsigned char* next  = (unsigned char*)(ws + 139520);    // 2097152 B

  setup_kernel<<<1, 256, 0, stream>>>(pa, ei, Am, stats);
  phase1_kernel<<<T_STEPS / 8, 256, 0, stream>>>(
      pa, times, g0, b0, W1, b1, g1, be1, W2, b2, g2, be2,
      W3, b3, g3, be3, W4, b4, Am, stats, next);
  chunkmap_kernel<<<64, 64, 0, stream>>>(next, cmap);
  scan_kernel<<<1, 64, 0, stream>>>(next, cmap, zone);
  phase3_kernel<<<T_STEPS / 16, 256, 0, stream>>>(
      pa, times, g0, b0, W1, b1, g1, be1, W2, b2, g2, be2,
      W3, b3, g3, be3, W4, b4, Am, stats, zone, out);
}